// SharedMGEmbedder_32667521253918
// MI455X (gfx1250) — compile-verified
//
#include <hip/hip_runtime.h>

// ---------------- problem constants (match reference) ----------------
#define B_   2
#define V_   4
#define F_   128
#define E_   128
#define NH_  7
#define N6_  196608
#define N5_  49152
#define N4_  12288
#define PF_  32768
#define P5_  (PF_ / 4)
#define P4_  (PF_ / 16)
#define O5_  4
#define O4_  16

// ---------------- tiling ----------------
#define WAVES         8
#define ROWS_PER_WAVE 16
#define TILE          (WAVES * ROWS_PER_WAVE)   // 128 output rows / workgroup
#define PW            130                        // Wt LDS pitch (floats)
#define PA            132                        // tile pitch (floats), rows 16B-aligned
#define SCR_ROWS      51                         // 16 A rows + (28 t5 + 7 t4) interp rows

#define LDS_WT_F    (E_ * PW)                    // 16640 floats
#define LDS_WV_F    (WAVES * 256)                // per-wave interp weights
#define LDS_SCR_F   (WAVES * SCR_ROWS * PA)      // per-wave gather tiles
#define LDS_TOTAL_F (LDS_WT_F + LDS_WV_F + LDS_SCR_F)
#define LDS_BYTES   (LDS_TOTAL_F * 4)            // ~283 KB < 320 KB/WG

typedef float v2f __attribute__((ext_vector_type(2)));
typedef float v8f __attribute__((ext_vector_type(8)));

// ---------------- CDNA5 async global->LDS path (guarded) ----------------
#if defined(__has_builtin)
#if __has_builtin(__builtin_amdgcn_global_load_async_to_lds_b128)
#define HAVE_ASYNC 1
#endif
#endif
#ifndef HAVE_ASYNC
#define HAVE_ASYNC 0
#endif

#if HAVE_ASYNC
#if __has_builtin(__builtin_amdgcn_s_wait_asynccnt)
#define WAIT_ASYNC(n) do { __builtin_amdgcn_s_wait_asynccnt(n); \
                           asm volatile("" ::: "memory"); } while (0)
#else
#define WAIT_ASYNC(n) asm volatile("s_wait_asynccnt " #n ::: "memory")
#endif
#else
#define WAIT_ASYNC(n) asm volatile("" ::: "memory")
#endif

// Builtin parameter types (from hipcc diagnostic): AS(1)/AS(3) pointers to
// GCC-vector int4. A v4i* converts implicitly to void* of the same AS too.
typedef int ai4 __attribute__((vector_size(4 * sizeof(int))));
typedef __attribute__((address_space(1))) ai4 gas_ai4;
typedef __attribute__((address_space(3))) ai4 las_ai4;

__device__ __forceinline__ void copy16_to_lds(const float* g, float* l) {
#if HAVE_ASYNC
  // global_load_async_to_lds_b128: per-lane 16B, tracked by ASYNCcnt
  __builtin_amdgcn_global_load_async_to_lds_b128(
      (gas_ai4*)(g), (las_ai4*)(l), 0, 0);
#else
  *(float4*)l = *(const float4*)g;
#endif
}

__global__ __launch_bounds__(WAVES * 32) void mg_embed_fused(
    const float* __restrict__ table6, const float* __restrict__ table5,
    const float* __restrict__ table4, const float* __restrict__ W,
    const float* __restrict__ bias,
    const float* __restrict__ rel5,  const float* __restrict__ rel4,
    const float* __restrict__ mask5, const float* __restrict__ mask4,
    const int* __restrict__ var_indices, const int* __restrict__ idx6,
    const int* __restrict__ nh_idx5, const int* __restrict__ nh_idx4,
    float* __restrict__ out)
{
  extern __shared__ float lds[];
  float* Wt  = lds;                          // [E_][PW] : Wt[e*PW + k] = W[k][e]
  float* wv  = lds + LDS_WT_F;               // per-wave: [16][8] w5 then [16][8] w4
  float* scr = lds + LDS_WT_F + LDS_WV_F;    // per-wave tiles, pitch PA

  const int lane = threadIdx.x & 31;
  const int wave = threadIdx.x >> 5;
  const int ll   = lane & 15;                // N column lane / A row
  const int lh   = lane >> 4;                // half-wave selector

  const int bv  = blockIdx.y;
  const int b   = bv / V_;
  const int pfw = blockIdx.x * TILE + wave * ROWS_PER_WAVE;   // this wave's 16 rows

  const int var = var_indices[bv];
  float* myA  = scr + wave * (SCR_ROWS * PA);   // 16 rows: GEMM A tile
  float* myI  = myA + 16 * PA;                  // 35 rows: interp tile
  float* wvw  = wv + wave * 256;

  // ---- Phase 1a: issue ALL row gathers (async -> LDS, ASYNCcnt-tracked) ----
  // First the 16 table6 rows (needed for the GEMM), then the 35 interp rows;
  // async loads complete in order, so ASYNCcnt<=35 means the A tile is ready.
  {
    const float* t6v = table6 + (size_t)var * N6_ * F_;
    const int*   i6  = idx6 + (size_t)b * PF_ + pfw;
    #pragma unroll 4
    for (int it = 0; it < ROWS_PER_WAVE; ++it) {
      const int row = i6[it];
      copy16_to_lds(t6v + (size_t)row * F_ + lane * 4, &myA[it * PA + lane * 4]);
    }
  }
  {
    const float* t5v = table5 + (size_t)var * N5_ * F_;
    const float* t4v = table4 + (size_t)var * N4_ * F_;
    const int* i5 = nh_idx5 + (size_t)b * (P5_ * NH_) + ((pfw >> 2) * NH_);
    const int* i4 = nh_idx4 + (size_t)b * (P4_ * NH_) + ((pfw >> 4) * NH_);
    #pragma unroll 7
    for (int it = 0; it < 28; ++it) {
      const int row = i5[it];                       // group g=it/7 laid out as g*7+n
      copy16_to_lds(t5v + (size_t)row * F_ + lane * 4, &myI[it * PA + lane * 4]);
    }
    #pragma unroll
    for (int n = 0; n < NH_; ++n) {
      const int row = i4[n];
      copy16_to_lds(t4v + (size_t)row * F_ + lane * 4, &myI[(28 + n) * PA + lane * 4]);
    }
  }

  // ---- Phase 0a (overlapped with async gathers): stage W transposed ----
  for (int i = threadIdx.x; i < F_ * E_; i += blockDim.x) {
    const int f = i >> 7, e = i & 127;
    Wt[e * PW + f] = W[i];
  }

  // ---- Phase 0b (overlapped): interpolation weights for this wave's rows ----
  {
    const int m  = ll;
    const int pf = pfw + m;
    if (lh == 0) {                                  // level-5 weights
      const int p5 = pf >> 2, o = pf & 3;
      const float* rp = rel5  + (((size_t)b * P5_ + p5) * O5_ + o) * NH_;
      const float* mp = mask5 + ((size_t)b * P5_ + p5) * NH_;
      float w[NH_]; float s = 0.f;
      #pragma unroll
      for (int n = 0; n < NH_; ++n) {
        const float rd = rp[n] + mp[n] * 1e10f;
        w[n] = 1.f / (1e-20f + rd);
        s += w[n];
      }
      const float inv = 1.f / s;
      #pragma unroll
      for (int n = 0; n < NH_; ++n) wvw[m * 8 + n] = w[n] * inv;
    } else {                                        // level-4 weights
      const int p4 = pf >> 4, o = pf & 15;
      const float* rp = rel4  + (((size_t)b * P4_ + p4) * O4_ + o) * NH_;
      const float* mp = mask4 + ((size_t)b * P4_ + p4) * NH_;
      float w[NH_]; float s = 0.f;
      #pragma unroll
      for (int n = 0; n < NH_; ++n) {
        const float rd = rp[n] + mp[n] * 1e10f;
        w[n] = 1.f / (1e-20f + rd);
        s += w[n];
      }
      const float inv = 1.f / s;
      #pragma unroll
      for (int n = 0; n < NH_; ++n) wvw[128 + m * 8 + n] = w[n] * inv;
    }
  }

  WAIT_ASYNC(35);      // A tile (first 16 async loads) resident in LDS
  __syncthreads();     // Wt + weights visible workgroup-wide

  // ---- Phase 2: C = bias; C += A(16x128) * W(128x128) via v_wmma_f32_16x16x4_f32 ----
  v8f acc[8];
  #pragma unroll
  for (int nt = 0; nt < 8; ++nt) {
    const float be = bias[nt * 16 + ll];
    acc[nt] = be;                                   // splat
  }
  const int koff = lh * 2;                          // lanes 16-31 hold K+2, K+3
  for (int k = 0; k < 32; ++k) {
    const v2f a = *(const v2f*)&myA[ll * PA + k * 4 + koff];
    #pragma unroll
    for (int nt = 0; nt < 8; ++nt) {
      const v2f bf = *(const v2f*)&Wt[(nt * 16 + ll) * PW + k * 4 + koff];
      acc[nt] = __builtin_amdgcn_wmma_f32_16x16x4_f32(
          false, a, false, bf, (short)0, acc[nt], false, false);
    }
  }

  WAIT_ASYNC(0);       // interp rows resident (gather latency hidden behind GEMM)

  // ---- Phase 4: add distance-weighted neighbor sums into the WMMA accumulators ----
  #pragma unroll
  for (int r = 0; r < 8; ++r) {
    const int m     = r + 8 * lh;                    // accumulator row per C/D layout
    const int base5 = (m >> 2) * 7 * PA;             // this row's p5 neighbor group
    const int wo    = m * 8;
    #pragma unroll
    for (int n = 0; n < NH_; ++n) {
      const float w5 = wvw[wo + n];
      const float w4 = wvw[128 + wo + n];
      const float* r5 = &myI[base5 + n * PA + ll];
      const float* r4 = &myI[28 * PA + n * PA + ll];
      #pragma unroll
      for (int nt = 0; nt < 8; ++nt) {
        acc[nt][r] = fmaf(w5, r5[nt * 16], fmaf(w4, r4[nt * 16], acc[nt][r]));
      }
    }
  }

  // ---- Phase 5: store (two coalesced 64B segments per instruction) ----
  float* op = out + ((size_t)bv * PF_ + pfw) * E_;
  #pragma unroll
  for (int nt = 0; nt < 8; ++nt) {
    #pragma unroll
    for (int r = 0; r < 8; ++r) {
      const int m = r + 8 * lh;
      op[(size_t)m * E_ + nt * 16 + ll] = acc[nt][r];
    }
  }
}

extern "C" void kernel_launch(void* const* d_in, const int* in_sizes, int n_in,
                              void* d_out, int out_size, void* d_ws, size_t ws_size,
                              hipStream_t stream) {
  (void)in_sizes; (void)n_in; (void)out_size; (void)d_ws; (void)ws_size;
  const float* table6 = (const float*)d_in[0];
  const float* table5 = (const float*)d_in[1];
  const float* table4 = (const float*)d_in[2];
  const float* W      = (const float*)d_in[3];
  const float* bias   = (const float*)d_in[4];
  const float* rel5   = (const float*)d_in[5];
  const float* rel4   = (const float*)d_in[6];
  const float* mask5  = (const float*)d_in[7];
  const float* mask4  = (const float*)d_in[8];
  const int*   vidx   = (const int*)d_in[9];
  const int*   idx6   = (const int*)d_in[10];
  const int*   nh5    = (const int*)d_in[11];
  const int*   nh4    = (const int*)d_in[12];
  float* out = (float*)d_out;

  dim3 grid(PF_ / TILE, B_ * V_);
  dim3 block(WAVES * 32);
  mg_embed_fused<<<grid, block, LDS_BYTES, stream>>>(
      table6, table5, table4, W, bias, rel5, rel4, mask5, mask4,
      vidx, idx6, nh5, nh4, out);
}